// EmbeddingLayer_34024730918990
// MI455X (gfx1250) — compile-verified
//
#include <hip/hip_runtime.h>
#include <math.h>

typedef __attribute__((ext_vector_type(2))) float v2f;
typedef __attribute__((ext_vector_type(8))) float v8f;

#define BATCH 4096
#define FEAT 64
#define EMB 32
#define KDIM 2048      // FEAT*EMB
#define MODELS 512
#define SEQ 128
#define KCHUNK 512
#define NCHUNKS (KDIM / KCHUNK)   // 4
#define LDSS (KCHUNK + 4)         // 516 floats: pad so row stride mod 64 == 4 -> conflict-free

// ---------------- Stage 0: positional encoding table [SEQ][MODELS] ----------------
__global__ __launch_bounds__(256) void pe_init_kernel(float* __restrict__ pe) {
  int idx = blockIdx.x * 256 + threadIdx.x;     // 0 .. SEQ*MODELS-1
  int m = idx & (MODELS - 1);
  int s = idx >> 9;                             // MODELS == 512
  float i2 = (float)((m >> 1) << 1);            // 2*floor(m/2)
  float rate = expf(-(i2 / (float)MODELS) * logf(10000.0f));
  float ang = (float)s * rate;
  pe[idx] = (m & 1) ? cosf(ang) : sinf(ang);
}

// ---------------- Stage 1: fused gather + GEMM via f32 WMMA ----------------
// Y[4096][512] = (emb[tokens] * sqrt(512)).reshape(4096,2048) @ W[2048][512] + bias
__global__ __launch_bounds__(256) void gemm_kernel(const int* __restrict__ tokens,
                                                   const float* __restrict__ emb,
                                                   const float* __restrict__ W,
                                                   const float* __restrict__ bias,
                                                   float* __restrict__ Y) {
  __shared__ float Xs[16 * LDSS];               // 16 rows x 516 floats ~= 33 KB

  const int mblk = blockIdx.x;                  // 0..255, 16 batch rows each
  const int tid  = threadIdx.x;
  const int lane = tid & 31;
  const int wave = tid >> 5;                    // 0..7
  const int lo   = lane & 15;                   // N column / A row within tile
  const int hi   = lane >> 4;                   // selects K pair {0,1} vs {2,3}
  const float scale = 22.627416997969522f;      // sqrt(512)

  v8f acc0 = {}, acc1 = {}, acc2 = {}, acc3 = {};

  const int gm = tid >> 4;                      // gather: batch row in tile, 0..15
  const int fl = tid & 15;                      // gather: local feature, 0..15

  for (int kc = 0; kc < NCHUNKS; ++kc) {
    __syncthreads();
    // Gather one K-chunk (16 features x 32 emb dims per row) into LDS, scaled.
    {
      int f   = kc * 16 + fl;
      int tok = tokens[(mblk * 16 + gm) * FEAT + f];
      const float4* src = (const float4*)(emb + (size_t)tok * EMB);
      float4* dst = (float4*)(Xs + gm * LDSS + fl * EMB);
#pragma unroll
      for (int q = 0; q < 8; ++q) {
        float4 v = src[q];
        v.x *= scale; v.y *= scale; v.z *= scale; v.w *= scale;
        dst[q] = v;
      }
    }
    __syncthreads();

    const int colbase = wave * 64;              // 4 N-tiles of 16 per wave
    for (int ks = 0; ks < KCHUNK; ks += 4) {
      // A 16x4 layout: lane -> M = lo; lanes 0-15 hold K={ks,ks+1}, lanes 16-31 K={ks+2,ks+3}
      v2f a = *(const v2f*)(Xs + lo * LDSS + ks + 2 * hi);
      int krow = kc * KCHUNK + ks + 2 * hi;     // global K row for this half-wave
      const float* wp = W + (size_t)krow * MODELS;
      {
        int col = colbase + lo;
        v2f b; b.x = wp[col]; b.y = wp[MODELS + col];
        acc0 = __builtin_amdgcn_wmma_f32_16x16x4_f32(false, a, false, b, (short)0, acc0, false, false);
      }
      {
        int col = colbase + 16 + lo;
        v2f b; b.x = wp[col]; b.y = wp[MODELS + col];
        acc1 = __builtin_amdgcn_wmma_f32_16x16x4_f32(false, a, false, b, (short)0, acc1, false, false);
      }
      {
        int col = colbase + 32 + lo;
        v2f b; b.x = wp[col]; b.y = wp[MODELS + col];
        acc2 = __builtin_amdgcn_wmma_f32_16x16x4_f32(false, a, false, b, (short)0, acc2, false, false);
      }
      {
        int col = colbase + 48 + lo;
        v2f b; b.x = wp[col]; b.y = wp[MODELS + col];
        acc3 = __builtin_amdgcn_wmma_f32_16x16x4_f32(false, a, false, b, (short)0, acc3, false, false);
      }
    }
  }

  // Store C/D: VGPR v holds M = v + 8*hi, N = lo (per ISA 16x16 f32 layout). Fold bias.
  int rowbase = mblk * 16 + 8 * hi;
#pragma unroll
  for (int j = 0; j < 4; ++j) {
    v8f acc = (j == 0) ? acc0 : (j == 1) ? acc1 : (j == 2) ? acc2 : acc3;
    int col = wave * 64 + j * 16 + lo;
    float bv = bias[col];
#pragma unroll
    for (int v = 0; v < 8; ++v) {
      Y[(size_t)(rowbase + v) * MODELS + col] = acc[v] + bv;
    }
  }
}

// ---------------- Stage 2: out[b][s][m] = Y[b][m] + PE[s][m], pure streaming ----------------
__global__ __launch_bounds__(256) void bcast_kernel(const float4* __restrict__ Y4,
                                                    const float4* __restrict__ PE4,
                                                    float4* __restrict__ out4) {
  size_t idx = (size_t)blockIdx.x * 256 + threadIdx.x;   // float4 index
  unsigned m4 = (unsigned)idx & 127u;                     // MODELS/4 == 128
  unsigned s  = ((unsigned)(idx >> 7)) & 127u;            // SEQ == 128
  unsigned b  = (unsigned)(idx >> 14);
  float4 y = Y4[(size_t)b * 128 + m4];
  float4 p = PE4[(size_t)s * 128 + m4];
  float4 o;
  o.x = y.x + p.x; o.y = y.y + p.y; o.z = y.z + p.z; o.w = y.w + p.w;
  out4[idx] = o;
}

extern "C" void kernel_launch(void* const* d_in, const int* in_sizes, int n_in,
                              void* d_out, int out_size, void* d_ws, size_t ws_size,
                              hipStream_t stream) {
  const int*   tokens = (const int*)d_in[0];
  const float* emb    = (const float*)d_in[1];
  const float* W      = (const float*)d_in[2];
  const float* bias   = (const float*)d_in[3];

  float* pe = (float*)d_ws;                                          // 128*512*4   = 256 KB
  float* Y  = (float*)((char*)d_ws + (size_t)SEQ * MODELS * sizeof(float)); // 4096*512*4 = 8 MB
  float* out = (float*)d_out;

  pe_init_kernel<<<(SEQ * MODELS) / 256, 256, 0, stream>>>(pe);
  gemm_kernel<<<BATCH / 16, 256, 0, stream>>>(tokens, emb, W, bias, Y);

  size_t total4 = (size_t)BATCH * SEQ * MODELS / 4;                  // 67,108,864 float4
  bcast_kernel<<<(unsigned)(total4 / 256), 256, 0, stream>>>(
      (const float4*)Y, (const float4*)pe, (float4*)out);
}